// RPN_53807350284714
// MI455X (gfx1250) — compile-verified
//
#include <hip/hip_runtime.h>
#include <cstdint>

// ---- constants from the reference ----
#define SCALE_CLAMP_F   3.3322045101752038f   // log(224/8)
#define NEG_BIG_F       (-100000000.0f)
#define MASK_THRESH_F   (-10000000.0f)

#define BLOCK    256
#define STAGES   4          // 4-deep async pipeline: 32 KB LDS / block (320 KB per WGP)
#define MAX_GRID 1600

// CDNA5 async global->LDS b128 load (GV mode: 64-bit vaddr in VGPR pair, saddr=off).
// %0 = VGPR with wave-relative LDS byte address, %1 = VGPR pair with 64-bit global addr.
__device__ __forceinline__ void async_ld_b128(const void* gaddr, unsigned lds_off) {
    asm volatile("global_load_async_to_lds_b128 %0, %1, off"
                 :: "v"(lds_off), "v"(gaddr)
                 : "memory");
}
__device__ __forceinline__ void wait_async6() { asm volatile("s_wait_asynccnt 6" ::: "memory"); }
__device__ __forceinline__ void wait_async4() { asm volatile("s_wait_asynccnt 4" ::: "memory"); }
__device__ __forceinline__ void wait_async2() { asm volatile("s_wait_asynccnt 2" ::: "memory"); }
__device__ __forceinline__ void wait_async0() { asm volatile("s_wait_asynccnt 0" ::: "memory"); }

__device__ __forceinline__ float4 decode_box(const float4 d, const float4 a) {
    const float dx = d.x, dy = d.y;
    const float dw = fminf(d.z, SCALE_CLAMP_F);
    const float dh = fminf(d.w, SCALE_CLAMP_F);

    const float pcx = (a.x + a.z) * 0.5f;
    const float pcy = (a.y + a.w) * 0.5f;
    const float pw  = a.z - a.x;
    const float ph  = a.w - a.y;

    const float bcx = fmaf(pw, dx, pcx);
    const float bcy = fmaf(ph, dy, pcy);
    const float hw  = pw * __expf(dw) * 0.5f;   // v_exp_f32 path
    const float hh  = ph * __expf(dh) * 0.5f;

    float4 r;
    r.x = (dx < MASK_THRESH_F) ? NEG_BIG_F : (bcx - hw);
    r.y = (dy < MASK_THRESH_F) ? NEG_BIG_F : (bcy - hh);
    r.z = (dw < MASK_THRESH_F) ? NEG_BIG_F : (bcx + hw);
    r.w = (dh < MASK_THRESH_F) ? NEG_BIG_F : (bcy + hh);
    return r;
}

__global__ __launch_bounds__(BLOCK)
void rpn_decode_async(const float4* __restrict__ deltas,
                      const float4* __restrict__ anchors,
                      float4* __restrict__ out,
                      int rows, int fullTiles,
                      int baseT, int remT)      // host-computed: fullTiles / grid, fullTiles % grid
{
    __shared__ float4 s_d[STAGES][BLOCK];
    __shared__ float4 s_a[STAGES][BLOCK];

    const int tid    = threadIdx.x;
    const int stride = gridDim.x;
    const int bid    = blockIdx.x;

    // Tiles this block owns (bid, bid+stride, ...): no device-side division.
    const int T = baseT + ((bid < remT) ? 1 : 0);

    // Wave-relative LDS byte offsets (low 32 bits of the flat address of an LDS
    // object are the wave-relative LDS offset). Stage k is at base + k*BLOCK*16.
    const unsigned ld_d_base = (unsigned)(uintptr_t)&s_d[0][tid];
    const unsigned ld_a_base = (unsigned)(uintptr_t)&s_a[0][tid];

    // Incremental per-lane global pointers (no per-iteration index multiplies).
    const long long step = (long long)stride * BLOCK;      // float4 elements
    const float4* gd = deltas  + (long long)bid * BLOCK + tid;
    const float4* ga = anchors + (long long)bid * BLOCK + tid;
    float4*       po = out     + (long long)bid * BLOCK + tid;

    // Prologue: stage up to 3 tiles ahead.
    #pragma unroll
    for (int p = 0; p < STAGES - 1; ++p) {
        if (p < T) {
            const unsigned so = (unsigned)p * (BLOCK * 16u);
            async_ld_b128(gd, ld_d_base + so);
            async_ld_b128(ga, ld_a_base + so);
            gd += step; ga += step;
        }
    }

    for (int j = 0; j < T; ++j) {
        if (j + (STAGES - 1) < T) {
            // Issue tile j+3 into its stage, then wait so tile j is resident:
            // tiles j..j+3 in flight = 8 loads -> wait asynccnt <= 6 (in-order completion).
            const unsigned so = (((unsigned)(j + STAGES - 1)) & (STAGES - 1u)) * (BLOCK * 16u);
            async_ld_b128(gd, ld_d_base + so);
            async_ld_b128(ga, ld_a_base + so);
            gd += step; ga += step;
            wait_async6();
        } else {
            const int ahead = T - 1 - j;                   // 0..2 in the drain
            if      (ahead == 2) wait_async4();
            else if (ahead == 1) wait_async2();
            else                 wait_async0();
        }

        const unsigned cur = (unsigned)j & (STAGES - 1u);
        const float4 d = s_d[cur][tid];                    // ds_load_b128
        const float4 a = s_a[cur][tid];                    // ds_load_b128

        *po = decode_box(d, a);                            // global_store_b128
        po += step;
    }

    // Remainder rows (rows % BLOCK, empty for N = 9600*256): plain guarded path.
    if (bid == 0) {
        const int r = fullTiles * BLOCK + tid;
        if (r < rows) {
            out[r] = decode_box(deltas[r], anchors[r]);
        }
    }
}

extern "C" void kernel_launch(void* const* d_in, const int* in_sizes, int n_in,
                              void* d_out, int out_size, void* d_ws, size_t ws_size,
                              hipStream_t stream) {
    (void)n_in; (void)out_size; (void)d_ws; (void)ws_size;
    const float4* deltas  = (const float4*)d_in[0];
    const float4* anchors = (const float4*)d_in[1];
    float4*       out     = (float4*)d_out;

    const int rows = in_sizes[0] / 4;          // (N,4) float32 flattened
    if (rows <= 0) return;
    const int fullTiles = rows / BLOCK;
    int grid = fullTiles < MAX_GRID ? fullTiles : MAX_GRID;
    if (grid < 1) grid = 1;                    // tail-only case still needs block 0

    const int baseT = fullTiles / grid;        // host-side division only
    const int remT  = fullTiles % grid;

    rpn_decode_async<<<grid, BLOCK, 0, stream>>>(deltas, anchors, out,
                                                 rows, fullTiles, baseT, remT);
}